// Layer_49615462203743
// MI455X (gfx1250) — compile-verified
//
#include <hip/hip_runtime.h>
#include <hip/hip_bf16.h>

#define N_NODES 10000
#define N_EDGES 640000
#define FEAT 128

typedef __attribute__((ext_vector_type(2))) float v2f;
typedef __attribute__((ext_vector_type(8))) float v8f;

// ---------------------------------------------------------------------------
// Kernel 0: zero the accumulation workspace (accp, accu, deg are contiguous)
// ---------------------------------------------------------------------------
__global__ void __launch_bounds__(256) zero_kernel(float* __restrict__ p, int n) {
    int i = blockIdx.x * 256 + threadIdx.x;
    if (i < n) p[i] = 0.0f;
}

// ---------------------------------------------------------------------------
// Kernel 1: edge aggregation.
// One wave (32 lanes) per edge; each lane handles 4 consecutive features.
// Gather h[src] (coalesced 512B/edge), atomic-accumulate h*e and h into the
// two per-node accumulators (both L2-resident, 5.1 MB each), count degree.
// ---------------------------------------------------------------------------
__global__ void __launch_bounds__(256)
edge_kernel(const float* __restrict__ h, const float* __restrict__ e,
            const int* __restrict__ src, const int* __restrict__ dst,
            float* __restrict__ accp, float* __restrict__ accu,
            float* __restrict__ deg) {
    int gid  = blockIdx.x * 256 + threadIdx.x;
    int edge = gid >> 5;
    int lane = gid & 31;
    if (edge >= N_EDGES) return;

    int s = src[edge];
    int d = dst[edge];
    float ev = e[edge];

    int c = lane << 2;  // 4 floats per lane
    const float4 hv = *(const float4*)(h + (size_t)s * FEAT + c);

    float* up = accu + (size_t)d * FEAT + c;
    float* pp = accp + (size_t)d * FEAT + c;
    atomicAdd(up + 0, hv.x);
    atomicAdd(up + 1, hv.y);
    atomicAdd(up + 2, hv.z);
    atomicAdd(up + 3, hv.w);
    atomicAdd(pp + 0, hv.x * ev);
    atomicAdd(pp + 1, hv.y * ev);
    atomicAdd(pp + 2, hv.z * ev);
    atomicAdd(pp + 3, hv.w * ev);
    if (lane == 0) atomicAdd(deg + d, 1.0f);
}

// ---------------------------------------------------------------------------
// WMMA fragment loader: row-major [rows][128] matrix, f32 16x16x4 A/B layout.
// lane L supplies element (row0 + (L&15), k0 + 2*(L>>4) .. +1) as a float2.
// ---------------------------------------------------------------------------
__device__ __forceinline__ v2f load_frag128(const float* __restrict__ base,
                                            int row, int k) {
    return *(const v2f*)(base + (size_t)row * FEAT + k);
}

// ---------------------------------------------------------------------------
// Kernel 2: GEMM1 — h_total = h@Ws^T + (accp/deg)@Wn^T + (accu/deg)@Wu^T + b
// One wave computes one 16x16 tile; grid = 625 blocks x 8 waves (mt x nt).
// All waves full -> EXEC all ones for every WMMA (ISA requirement).
// ---------------------------------------------------------------------------
__global__ void __launch_bounds__(256)
gemm1_kernel(const float* __restrict__ h,
             const float* __restrict__ accp,
             const float* __restrict__ accu,
             const float* __restrict__ deg,
             const float* __restrict__ Ws, const float* __restrict__ Wsb,
             const float* __restrict__ Wn, const float* __restrict__ Wnb,
             const float* __restrict__ Wu, const float* __restrict__ Wub,
             float* __restrict__ htot) {
    const int wave = threadIdx.x >> 5;
    const int lane = threadIdx.x & 31;
    const int mt = blockIdx.x;      // 0..624
    const int nt = wave;            // 0..7
    const int r    = lane & 15;
    const int koff = (lane >> 4) << 1;
    const int mrow = mt * 16 + r;   // A row this lane feeds
    const int nrow = nt * 16 + r;   // B (= weight) row this lane feeds

    const float id = 1.0f / fmaxf(deg[mrow], 1.0f);  // mean normalization

    v8f acc = {};
    for (int kk = 0; kk < FEAT; kk += 4) {
        const int k = kk + koff;
        v2f a0 = load_frag128(h,    mrow, k);
        v2f a1 = load_frag128(accp, mrow, k) * id;
        v2f a2 = load_frag128(accu, mrow, k) * id;
        v2f b0 = load_frag128(Ws, nrow, k);
        v2f b1 = load_frag128(Wn, nrow, k);
        v2f b2 = load_frag128(Wu, nrow, k);
        acc = __builtin_amdgcn_wmma_f32_16x16x4_f32(false, a0, false, b0,
                                                    (short)0, acc, false, false);
        acc = __builtin_amdgcn_wmma_f32_16x16x4_f32(false, a1, false, b1,
                                                    (short)0, acc, false, false);
        acc = __builtin_amdgcn_wmma_f32_16x16x4_f32(false, a2, false, b2,
                                                    (short)0, acc, false, false);
    }

    const int ncol = nt * 16 + r;                       // N = lane&15
    const float bias = Wsb[ncol] + Wnb[ncol] + Wub[ncol];
    const int mbase = mt * 16 + ((lane >> 4) << 3);     // M = v + 8*(lane>>4)
#pragma unroll
    for (int v = 0; v < 8; ++v)
        htot[(size_t)(mbase + v) * FEAT + ncol] = acc[v] + bias;
}

// ---------------------------------------------------------------------------
// Kernel 3: GEMM2 — out = h_total @ lin_w^T + lin_b
// ---------------------------------------------------------------------------
__global__ void __launch_bounds__(256)
gemm2_kernel(const float* __restrict__ htot,
             const float* __restrict__ lw, const float* __restrict__ lb,
             float* __restrict__ out) {
    const int wave = threadIdx.x >> 5;
    const int lane = threadIdx.x & 31;
    const int mt = blockIdx.x;
    const int nt = wave;
    const int r    = lane & 15;
    const int koff = (lane >> 4) << 1;
    const int mrow = mt * 16 + r;
    const int nrow = nt * 16 + r;

    v8f acc = {};
    for (int kk = 0; kk < FEAT; kk += 4) {
        const int k = kk + koff;
        v2f a = load_frag128(htot, mrow, k);
        v2f b = load_frag128(lw,   nrow, k);
        acc = __builtin_amdgcn_wmma_f32_16x16x4_f32(false, a, false, b,
                                                    (short)0, acc, false, false);
    }

    const int ncol = nt * 16 + r;
    const float bias = lb[ncol];
    const int mbase = mt * 16 + ((lane >> 4) << 3);
#pragma unroll
    for (int v = 0; v < 8; ++v)
        out[(size_t)(mbase + v) * FEAT + ncol] = acc[v] + bias;
}

// ---------------------------------------------------------------------------
extern "C" void kernel_launch(void* const* d_in, const int* in_sizes, int n_in,
                              void* d_out, int out_size, void* d_ws, size_t ws_size,
                              hipStream_t stream) {
    const float* h   = (const float*)d_in[0];
    const float* e   = (const float*)d_in[1];
    const int*   src = (const int*)d_in[2];
    const int*   dst = (const int*)d_in[3];
    const float* Ws  = (const float*)d_in[4];
    const float* Wsb = (const float*)d_in[5];
    const float* Wn  = (const float*)d_in[6];
    const float* Wnb = (const float*)d_in[7];
    const float* Wu  = (const float*)d_in[8];
    const float* Wub = (const float*)d_in[9];
    const float* lw  = (const float*)d_in[10];
    const float* lb  = (const float*)d_in[11];
    float* out = (float*)d_out;

    // workspace layout (floats)
    float* ws   = (float*)d_ws;
    float* accp = ws;                              // N_NODES*FEAT
    float* accu = ws + (size_t)N_NODES * FEAT;     // N_NODES*FEAT
    float* deg  = ws + (size_t)2 * N_NODES * FEAT; // N_NODES
    float* htot = deg + N_NODES;                   // N_NODES*FEAT

    const int nzero = 2 * N_NODES * FEAT + N_NODES;  // accp+accu+deg
    zero_kernel<<<(nzero + 255) / 256, 256, 0, stream>>>(ws, nzero);

    const int edge_threads = N_EDGES * 32;
    edge_kernel<<<edge_threads / 256, 256, 0, stream>>>(h, e, src, dst,
                                                        accp, accu, deg);

    // 10000 rows = 625 tiles of 16; 128 cols = 8 tiles of 16 (one per wave)
    gemm1_kernel<<<N_NODES / 16, 256, 0, stream>>>(h, accp, accu, deg,
                                                   Ws, Wsb, Wn, Wnb, Wu, Wub,
                                                   htot);
    gemm2_kernel<<<N_NODES / 16, 256, 0, stream>>>(htot, lw, lb, out);
}